// spuLayerTransformer_17824114278567
// MI455X (gfx1250) — compile-verified
//
#include <hip/hip_runtime.h>
#include <hip/hip_bf16.h>

// ---------------------------------------------------------------------------
// SPU bound-propagation: elementwise over d = 16.7M float32.
// Memory-bound streamer: 24 B/elem -> ~402 MB -> ~17.3 us floor @ 23.3 TB/s.
// Strategy: b128 non-temporal loads/stores, grid-stride + global_prefetch,
// ALL transcendental/reciprocal work on the TRANS pipe (v_exp_f32/v_rcp_f32)
// so it co-executes with VALU; one sigmoid per evaluation point (4/element).
// ---------------------------------------------------------------------------

typedef __attribute__((ext_vector_type(4))) float v4f;

// sigmoid(-z) = 1 / (1 + exp(z)); v_exp_f32 + v_rcp_f32 (~1-2 ulp).
// Saturates correctly: z -> +inf => exp -> inf => rcp -> 0 ; z -> -inf => 1.
__device__ __forceinline__ float sigmoid_neg(float z) {
    float e = __expf(z);                      // v_mul + v_exp_f32 (TRANS)
    return __builtin_amdgcn_rcpf(1.0f + e);   // v_add + v_rcp_f32 (TRANS)
}

// spu(z) with the sigmoid value hoisted so it can be shared with dspu.
__device__ __forceinline__ float spu_with_s(float z, float s) {
    return (z >= 0.0f) ? (z * z - 0.5f) : (s - 1.0f);
}

__device__ __forceinline__ void spu_bounds(float x, float l, float u,
                                           float &xo, float &lo, float &uo) {
    // p = clip(u - (|l| + |u|) / 2, l, u)
    float p = u - (fabsf(l) + fabsf(u)) * 0.5f;
    p = fminf(fmaxf(p, l), u);

    // One sigmoid per evaluation point: 4 exp + 4 rcp per element, all TRANS.
    float s_l = sigmoid_neg(l);
    float s_u = sigmoid_neg(u);
    float s_p = sigmoid_neg(p);
    float s_x = sigmoid_neg(x);

    float sl = spu_with_s(l, s_l);
    float su = spu_with_s(u, s_u);
    float sp = spu_with_s(p, s_p);

    // chord through (l, sl) and (u, su) — rcp instead of precise div:
    // keeps the ~10-op IEEE div sequence off the VALU pipe.
    float chord_slope = (su - sl) *
        __builtin_amdgcn_rcpf((u != l) ? (u - l) : 1.0f);
    float chord_int   = sl - chord_slope * l;

    // tangent at p (shares s_p with sp)
    float tan_slope = (p >= 0.0f) ? (2.0f * p) : (-s_p * (1.0f - s_p));
    float tan_int   = sp - p * tan_slope;

    // line through (l, sl) and (0, -0.5)
    float zl_slope = (-0.5f - sl) *
        __builtin_amdgcn_rcpf((l != 0.0f) ? -l : 1.0f);

    bool upos = (u > 0.0f);
    bool ppos = (p >= 0.0f);

    float lb_slope = upos ? (ppos ? tan_slope : zl_slope) : chord_slope;
    float lb_int   = upos ? (ppos ? tan_int   : -0.5f   ) : chord_int;
    float ub_slope = upos ? chord_slope : tan_slope;
    float ub_int   = upos ? chord_int   : tan_int;

    lo = lb_int + ((lb_slope > 0.0f) ? l : u) * lb_slope;
    uo = ub_int + ((ub_slope > 0.0f) ? u : l) * ub_slope;
    xo = spu_with_s(x, s_x);
}

__global__ __launch_bounds__(256)
void spu_stream_kernel(const float* __restrict__ x,
                       const float* __restrict__ l,
                       const float* __restrict__ u,
                       float* __restrict__ ox,
                       float* __restrict__ ol,
                       float* __restrict__ ou,
                       int n4, int rem) {
    const int tid    = blockIdx.x * blockDim.x + threadIdx.x;
    const int stride = gridDim.x * blockDim.x;

    const v4f* __restrict__ x4 = (const v4f*)x;
    const v4f* __restrict__ l4 = (const v4f*)l;
    const v4f* __restrict__ u4 = (const v4f*)u;
    v4f* __restrict__ ox4 = (v4f*)ox;
    v4f* __restrict__ ol4 = (v4f*)ol;
    v4f* __restrict__ ou4 = (v4f*)ou;

    for (int i = tid; i < n4; i += stride) {
        // gfx1250 prefetch path: pull next grid-stride iteration's lines.
        // Whole-machine footprint between prefetch and use (~48 MB across
        // 3 arrays) fits in the 192 MB L2.
        if (i + stride < n4) {
            __builtin_prefetch(&x4[i + stride], 0, 1);  // global_prefetch_b8
            __builtin_prefetch(&l4[i + stride], 0, 1);
            __builtin_prefetch(&u4[i + stride], 0, 1);
        }

        // Single-use stream: non-temporal b128 loads (th:TH_NT).
        v4f xv = __builtin_nontemporal_load(&x4[i]);
        v4f lv = __builtin_nontemporal_load(&l4[i]);
        v4f uv = __builtin_nontemporal_load(&u4[i]);

        v4f xo, lo, uo;
#pragma unroll
        for (int k = 0; k < 4; ++k) {
            float a, b, c;
            spu_bounds(xv[k], lv[k], uv[k], a, b, c);
            xo[k] = a; lo[k] = b; uo[k] = c;
        }

        // Non-temporal b128 stores: don't cache the write-once output stream.
        __builtin_nontemporal_store(xo, &ox4[i]);
        __builtin_nontemporal_store(lo, &ol4[i]);
        __builtin_nontemporal_store(uo, &ou4[i]);
    }

    // Scalar tail (n % 4 elements). n = 16,777,216 -> rem == 0, but keep
    // the kernel correct for any n.
    if (tid < rem) {
        int j = n4 * 4 + tid;
        float a, b, c;
        spu_bounds(x[j], l[j], u[j], a, b, c);
        ox[j] = a; ol[j] = b; ou[j] = c;
    }
}

extern "C" void kernel_launch(void* const* d_in, const int* in_sizes, int n_in,
                              void* d_out, int out_size, void* d_ws, size_t ws_size,
                              hipStream_t stream) {
    (void)n_in; (void)d_ws; (void)ws_size; (void)out_size;

    const float* x = (const float*)d_in[0];
    const float* l = (const float*)d_in[1];
    const float* u = (const float*)d_in[2];
    float* out = (float*)d_out;

    const int n  = in_sizes[0];          // 16,777,216
    const int n4 = n >> 2;
    const int rem = n & 3;

    float* ox = out;                     // x_out
    float* ol = out + (size_t)n;         // l_out
    float* ou = out + 2 * (size_t)n;     // u_out

    const int threads = 256;             // 8 wave32 per block
    long work = (n4 > 0) ? (long)n4 : (long)rem;
    int blocks = (int)((work + threads - 1) / threads);
    if (blocks > 4096) blocks = 4096;    // grid-stride: persistent waves,
    if (blocks < 1)    blocks = 1;       // makes the prefetch distance useful

    spu_stream_kernel<<<blocks, threads, 0, stream>>>(x, l, u, ox, ol, ou, n4, rem);
}